// IF_diff_18622978195598
// MI455X (gfx1250) — compile-verified
//
#include <hip/hip_runtime.h>

// IF (integrate-and-fire) temporal scan, T=8, forward only.
// Layout: x/out are [(T+1)*B, D] row-major; slice stride S = B*D = 4,194,304 f32.
// Memory-bound streaming kernel (~302 MB total traffic -> ~13 us at 23.3 TB/s):
// b128 everywhere, CDNA5 async global->LDS staging (ASYNCcnt), NT store hints
// (write-once output). WMMA is inapplicable: elementwise temporal scan, no
// contraction dimension.

#define T_STEPS 8

typedef float v4f __attribute__((ext_vector_type(4)));
typedef int   v4i __attribute__((ext_vector_type(4)));

#define AS1 __attribute__((address_space(1)))
#define AS3 __attribute__((address_space(3)))

#if defined(__AMDGCN__) && __has_builtin(__builtin_amdgcn_global_load_async_to_lds_b128)
#define IF_ASYNC_LDS 1
#else
#define IF_ASYNC_LDS 0
#endif

__device__ __forceinline__ v4f splat4(float s) {
    v4f r = {s, s, s, s};
    return r;
}

__global__ __launch_bounds__(256) void if_scan_kernel(
    const float* __restrict__ x, const float* __restrict__ thresh_p,
    float* __restrict__ out, long long S4 /* slice stride in float4 units */) {
    const long long c4 = (long long)blockIdx.x * 256 + threadIdx.x;  // float4 column id
    if (c4 >= S4) return;

    const float th = thresh_p[0];  // uniform scalar load
    const v4f* __restrict__ gsrc = (const v4f*)x + c4;
    v4f* __restrict__ gdst = (v4f*)out + c4;

#if IF_ASYNC_LDS
    // Per-wave private staging region: 9 slices * 32 lanes * 16B = 4.5 KB/wave,
    // 36 KB per 256-thread block (8 blocks/WGP by LDS -> 64 waves/WGP). Each
    // wave stages only data it consumes -> no barrier, just s_wait_asynccnt
    // before the ds reads.
    __shared__ v4f tile[(T_STEPS + 1) * 8 * 32];
    const int lane = threadIdx.x & 31;
    const int wave = threadIdx.x >> 5;
    v4f* my = &tile[(wave * (T_STEPS + 1)) * 32 + lane];
#pragma unroll
    for (int t = 0; t <= T_STEPS; ++t) {
        __builtin_amdgcn_global_load_async_to_lds_b128(
            (AS1 v4i*)(gsrc + (long long)t * S4),
            (AS3 v4i*)(my + t * 32),
            /*offset=*/0, /*cpol=*/0);
    }
    asm volatile("s_wait_asynccnt 0" ::: "memory");
#define XT(t) (my[(t)*32])
#else
    // Fallback: 9 outstanding non-temporal b128 loads into registers.
    v4f xv[T_STEPS + 1];
#pragma unroll
    for (int t = 0; t <= T_STEPS; ++t)
        xv[t] = __builtin_nontemporal_load(gsrc + (long long)t * S4);
#define XT(t) (xv[(t)])
#endif

    // 1/t; exact for t in {1,2,4,8}, <=1 ulp elsewhere (avoids fp32 div sequence).
    const float rcp_t[T_STEPS + 1] = {0.f,       1.f,       0.5f,
                                      1.f / 3.f, 0.25f,     0.2f,
                                      1.f / 6.f, 1.f / 7.f, 0.125f};

    v4f x0 = XT(0);
    v4f mem = splat4(0.5f * th);
    v4f exp_in = x0;
    v4f exp_out = splat4(0.f);
    v4f cum = splat4(0.f);

    // out[0] slice is zeros (harness poisons d_out, so write it).
    __builtin_nontemporal_store(splat4(0.f), gdst);

#pragma unroll
    for (int t = 1; t <= T_STEPS; ++t) {
        v4f xt = XT(t);
        v4f spike;
#pragma unroll
        for (int k = 0; k < 4; ++k) {
            float d = xt[k] - x0[k];
            float m = mem[k] + d + exp_in[k] - exp_out[k];
            float c = cum[k] + exp_out[k];
            // spike = zif(mem - thresh) * thresh
            float s = ((m - th) >= 0.f) ? th : 0.f;
            // spike -= zif(-mem) * gate * thresh.
            // Both factors are exactly 0 or 1, so
            // zif(-m)*gate*th == ((-m >= 0) && (c - th >= 0)) ? th : 0 exactly.
            // Fusing turns 2 cndmask + 2 mul into a VCC-mask AND + 1 cndmask.
            bool neg_fire = ((-m) >= 0.f) & ((c - th) >= 0.f);
            s -= neg_fire ? th : 0.f;
            c += s;
            m -= s;
            exp_in[k] += d * rcp_t[t];
            exp_out[k] += s * rcp_t[t];
            mem[k] = m;
            cum[k] = c;
            spike[k] = s;
        }
        __builtin_nontemporal_store(spike, gdst + (long long)t * S4);
    }
#undef XT
}

extern "C" void kernel_launch(void* const* d_in, const int* in_sizes, int n_in,
                              void* d_out, int out_size, void* d_ws, size_t ws_size,
                              hipStream_t stream) {
    const float* x = (const float*)d_in[0];
    const float* thresh = (const float*)d_in[1];
    float* out = (float*)d_out;

    long long n = (long long)in_sizes[0];        // (T+1) * B * D
    long long S = n / (T_STEPS + 1);             // slice stride in floats (B*D)
    long long S4 = S >> 2;                       // in float4 units (D=65536 -> divisible)

    int blocks = (int)((S4 + 255) / 256);
    if_scan_kernel<<<blocks, 256, 0, stream>>>(x, thresh, out, S4);
}